// MotionMamba_88699664597217
// MI455X (gfx1250) — compile-verified
//
#include <hip/hip_runtime.h>
#include <math.h>

// ---------------------------------------------------------------------------
// MotionMamba for MI455X (gfx1250, wave32). All GEMM-shaped compute uses
// V_WMMA_F32_16X16X4_F32 (fp32 matches the fp32 reference; problem is
// latency-bound, ~30 GFLOP, intermediates ~100MB < 192MB L2).
// Round 3: branchless K-loops. All reduction depths are multiples of 4, so
// k-guards are removed; out-of-range M/N rows are CLAMPED (their products
// land only in accumulator entries the epilogue never stores), making every
// inner-loop load unconditional (b64 where pairs are adjacent); im2col
// padding handled with cndmask selects, not exec-mask branches. EXEC stays
// all-ones into the v_wmma ops as required by the ISA.
// ---------------------------------------------------------------------------

typedef __attribute__((ext_vector_type(2))) float v2f;
typedef __attribute__((ext_vector_type(8))) float v8f;

__constant__ int c_index[52] = {
    0, 2, 4, 6, 8, 10, 12, 14, 16, 18, 20, 21, 22, 23, 24, 26, 28, 29,
    30, 31, 32, 33, 34, 36, 38, 39, 40, 41, 42, 44, 46, 47, 48, 49, 50,
    51, 52, 56, 57, 58, 59, 60, 62, 64, 66, 68, 70, 72, 74, 76, 78, 80};

// act: 0=none 1=silu 2=softshrink(0.5) 3=softplus
__device__ __forceinline__ float act_apply(float x, int mode) {
  if (mode == 1) return x * (1.0f / (1.0f + __expf(-x)));
  if (mode == 2) {
    float a = fmaxf(fabsf(x) - 0.5f, 0.0f);
    float s = (x > 0.0f) ? 1.0f : ((x < 0.0f) ? -1.0f : 0.0f);
    return s * a;
  }
  if (mode == 3) return (x > 20.0f) ? x : log1pf(__expf(x));
  return x;
}

__device__ __forceinline__ v8f wmma_f32_k4(v8f c, float a0, float a1, float b0,
                                           float b1) {
  v2f av = {a0, a1};
  v2f bv = {b0, b1};
  // D = A(16x4,f32) * B(4x16,f32) + C(16x16,f32)
  return __builtin_amdgcn_wmma_f32_16x16x4_f32(false, av, false, bv, (short)0,
                                               c, false, false);
}

__device__ __forceinline__ v2f ld2(const float* p) {
  return *(const v2f*)p;  // 8B-aligned by construction (even row strides)
}

// ---------------------------------------------------------------------------
// Implicit-GEMM conv, templated on kernel size (compile-time divisors in the
// im2col decode). One wave computes a 16(out-ch) x 32(pixel) strip = two
// 16x16 WMMA tiles sharing the A (weight) fragment. Requires Kred % 4 == 0.
// ---------------------------------------------------------------------------
template <int KSZ>
__global__ __launch_bounds__(128) void conv_wmma_kernel(
    const float* __restrict__ in1, const float* __restrict__ in2, int c_split,
    const float* __restrict__ wgt, const float* __restrict__ gsc,
    const float* __restrict__ bias, const float* __restrict__ resid,
    float* __restrict__ out, int B, int Cin, int Cout, int H, int W,
    int groups, int act) {
  constexpr int PAD = KSZ / 2;
  constexpr int KK2 = KSZ * KSZ;
  const int HW = H * W;
  const int cin_g = Cin / groups;
  const int cout_g = Cout / groups;
  const int Kred = cin_g * KK2;  // multiple of 4 for all layers used
  const int mtiles_g = (cout_g + 15) >> 4;

  const int lane = threadIdx.x & 31;
  const int wv = threadIdx.x >> 5;
  const int strip = blockIdx.x * (blockDim.x >> 5) + wv;  // 32-pixel strip
  const int g = blockIdx.y / mtiles_g;
  const int mt = blockIdx.y % mtiles_g;
  const int b = blockIdx.z;

  const int half = lane >> 4;  // 0 | 1
  const int mrow = lane & 15;  // A row (m) and B col (pixel n)
  const int koff = half * 2;
  const int c2in = Cin - c_split;

  // pixel decode for both tiles, hoisted out of the K loop
  const int p0 = strip * 32 + mrow;
  const int p1 = p0 + 16;
  const int p0c = (p0 < HW) ? p0 : (HW - 1);
  const int p1c = (p1 < HW) ? p1 : (HW - 1);
  const int py0 = p0c / W, px0 = p0c - py0 * W;
  const int py1 = p1c / W, px1 = p1c - py1 * W;

  // clamp OOB output rows: their products only reach unstored C entries
  const int ma = mt * 16 + mrow;
  const int mac = (ma < cout_g) ? ma : (cout_g - 1);
  const float* wrow = wgt + (long)(g * cout_g + mac) * Kred + koff;

  v8f acc0 = {};
  v8f acc1 = {};
  for (int kk = 0; kk < Kred; kk += 4) {
    // ---- A fragment: weights (shared by both N tiles), unconditional b64 ----
    v2f a = ld2(wrow + kk);
    __builtin_prefetch(wrow + kk + 64, 0, 1);  // global_prefetch_b8
    // ---- B fragments: im2col patches, branchless (clamped addr + select) ---
    float b0[2], b1[2];
#pragma unroll
    for (int q = 0; q < 2; ++q) {
      int kidx = kk + koff + q;
      int ci = kidx / KK2;  // compile-time divisor
      int r = kidx - ci * KK2;
      int ky = r / KSZ, kx = r - ky * KSZ;  // compile-time divisor
      int cg = g * cin_g + ci;
      bool use1 = (cg < c_split);
      const float* basep = use1 ? in1 : in2;
      long choff = use1 ? (((long)b * c_split + cg) * HW)
                        : (((long)b * c2in + (cg - c_split)) * HW);
      const float* base = basep + choff;
      {
        int iy = py0 + ky - PAD, ix = px0 + kx - PAD;
        bool ok = (p0 < HW) & (iy >= 0) & (iy < H) & (ix >= 0) & (ix < W);
        const float* ad = ok ? (base + (long)iy * W + ix) : in1;
        float v = *ad;
        b0[q] = ok ? v : 0.f;
      }
      {
        int iy = py1 + ky - PAD, ix = px1 + kx - PAD;
        bool ok = (p1 < HW) & (iy >= 0) & (iy < H) & (ix >= 0) & (ix < W);
        const float* ad = ok ? (base + (long)iy * W + ix) : in1;
        float v = *ad;
        b1[q] = ok ? v : 0.f;
      }
    }
    acc0 = wmma_f32_k4(acc0, a.x, a.y, b0[0], b0[1]);
    acc1 = wmma_f32_k4(acc1, a.x, a.y, b1[0], b1[1]);
  }

// ---- epilogue: C VGPR i -> rows m = mt*16 + i + half*8, col n = mrow ----
#pragma unroll
  for (int i = 0; i < 8; ++i) {
    int m = mt * 16 + i + half * 8;
    if (m < cout_g) {
      int gm = g * cout_g + m;
      float gv = gsc ? gsc[gm] : 1.0f;
      float bv = bias ? bias[gm] : 0.0f;
      if (p0 < HW) {
        float v = act_apply(acc0[i] * gv + bv, act);
        long o = ((long)b * Cout + gm) * HW + p0;
        if (resid) v += resid[o];
        out[o] = v;
      }
      if (p1 < HW) {
        float v = act_apply(acc1[i] * gv + bv, act);
        long o = ((long)b * Cout + gm) * HW + p1;
        if (resid) v += resid[o];
        out[o] = v;
      }
    }
  }
}

// ---------------------------------------------------------------------------
// GEMM: out[n, m] = sum_k A[m, k] * Bm[n, k]  (+bias[m], act)
// A row-major (M,K); Bm row-major (N, ldb); out row-major (N, ldo).
// One wave computes two 16x16 tiles along N sharing the A fragment.
// Requires K % 4 == 0 and even ldb (true for 64/128/36/4 used here).
// Inner loop: 3 unconditional b64 loads + 2 v_wmma, no branches.
// ---------------------------------------------------------------------------
__global__ __launch_bounds__(128) void gemm_tn_wmma_kernel(
    const float* __restrict__ A, const float* __restrict__ Bm,
    float* __restrict__ out, int M, long N, int K, int ldb, int ldo,
    const float* __restrict__ bias, int act) {
  const int lane = threadIdx.x & 31;
  const int wv = threadIdx.x >> 5;
  const long strip = (long)blockIdx.x * (blockDim.x >> 5) + wv;
  const int mt = blockIdx.y;
  const int half = lane >> 4;
  const int row = lane & 15;

  const long n0 = strip * 32 + row;
  const long n1 = n0 + 16;
  const int ma = mt * 16 + row;
  // clamp OOB rows/cols: their products only reach unstored C entries
  const float* ap = A + (long)((ma < M) ? ma : (M - 1)) * K + half * 2;
  const float* bp0 = Bm + ((n0 < N) ? n0 : (N - 1)) * ldb + half * 2;
  const float* bp1 = Bm + ((n1 < N) ? n1 : (N - 1)) * ldb + half * 2;

  v8f acc0 = {};
  v8f acc1 = {};
  for (int kk = 0; kk < K; kk += 4) {
    v2f a = ld2(ap + kk);
    v2f b0 = ld2(bp0 + kk);
    v2f b1 = ld2(bp1 + kk);
    acc0 = wmma_f32_k4(acc0, a.x, a.y, b0.x, b0.y);
    acc1 = wmma_f32_k4(acc1, a.x, a.y, b1.x, b1.y);
  }
#pragma unroll
  for (int i = 0; i < 8; ++i) {
    int m = mt * 16 + i + half * 8;
    if (m < M) {
      float bv = bias ? bias[m] : 0.0f;
      if (n0 < N) out[n0 * ldo + m] = act_apply(acc0[i] + bv, act);
      if (n1 < N) out[n1 * ldo + m] = act_apply(acc1[i] + bv, act);
    }
  }
}

// ---------------------------------------------------------------------------
// Correlation: out[b, j, y, x] = (1/C) * sum_c f1[b,c,y,x]*f2[b,c,y+dy,x+dx]
// ---------------------------------------------------------------------------
__global__ void correlation_kernel(const float* __restrict__ f1,
                                   const float* __restrict__ f2,
                                   float* __restrict__ out, int B, int C, int H,
                                   int W) {
  long gid = (long)blockIdx.x * blockDim.x + threadIdx.x;
  long total = (long)B * 52 * H * W;
  if (gid >= total) return;
  int x = gid % W;
  int y = (gid / W) % H;
  int j = (gid / ((long)W * H)) % 52;
  int b = gid / ((long)52 * H * W);
  int idx = c_index[j];
  int dy = idx / 9 - 4, dx = idx % 9 - 4;
  int y2 = y + dy, x2 = x + dx;
  float acc = 0.f;
  if (y2 >= 0 && y2 < H && x2 >= 0 && x2 < W) {
    const float* p1 = f1 + (long)b * C * H * W + (long)y * W + x;
    const float* p2 = f2 + (long)b * C * H * W + (long)y2 * W + x2;
    long st = (long)H * W;
    for (int c = 0; c < C; ++c) acc += p1[c * st] * p2[c * st];
  }
  out[gid] = acc / (float)C;
}

// bilinear 2x upsample (half-pixel centers, edge clamp)
__global__ void up2_kernel(const float* __restrict__ in, float* __restrict__ out,
                           int BC, int H, int W) {
  int OH = 2 * H, OW = 2 * W;
  long gid = (long)blockIdx.x * blockDim.x + threadIdx.x;
  long total = (long)BC * OH * OW;
  if (gid >= total) return;
  int ox = gid % OW;
  int oy = (gid / OW) % OH;
  long bc = gid / ((long)OW * OH);
  float fy = (oy + 0.5f) * 0.5f - 0.5f;
  float fx = (ox + 0.5f) * 0.5f - 0.5f;
  int y0 = (int)floorf(fy), x0 = (int)floorf(fx);
  float wy = fy - y0, wx = fx - x0;
  int y1 = min(max(y0 + 1, 0), H - 1), x1 = min(max(x0 + 1, 0), W - 1);
  y0 = min(max(y0, 0), H - 1);
  x0 = min(max(x0, 0), W - 1);
  const float* pp = in + bc * (long)H * W;
  float v = (1.f - wy) * ((1.f - wx) * pp[(long)y0 * W + x0] +
                          wx * pp[(long)y0 * W + x1]) +
            wy * ((1.f - wx) * pp[(long)y1 * W + x0] +
                  wx * pp[(long)y1 * W + x1]);
  out[gid] = v;
}

// causal depthwise conv1d (k=4, left pad 3) + silu; xz row = [xi(128) z(128)]
__global__ void dwconv1d_silu_kernel(const float* __restrict__ xz,
                                     const float* __restrict__ w,
                                     const float* __restrict__ bias,
                                     float* __restrict__ xc, int Bseq, int L) {
  long gid = (long)blockIdx.x * blockDim.x + threadIdx.x;
  long total = (long)Bseq * L * 128;
  if (gid >= total) return;
  int d = gid & 127;
  int l = (gid >> 7) % L;
  long b = gid / ((long)128 * L);
  float acc = bias[d];
#pragma unroll
  for (int j = 0; j < 4; ++j) {
    int li = l - 3 + j;
    float xv = (li >= 0) ? xz[(b * L + li) * 256 + d] : 0.f;
    acc = fmaf(w[d * 4 + j], xv, acc);
  }
  xc[gid] = acc * (1.0f / (1.0f + __expf(-acc)));
}

// selective scan: one block per sequence, 128 threads = channels; B_t/C_t in LDS
__global__ __launch_bounds__(128) void mamba_scan_kernel(
    const float* __restrict__ dtb, const float* __restrict__ xc,
    const float* __restrict__ xdbl, const float* __restrict__ A_log,
    float* __restrict__ ys, int L) {
  int b = blockIdx.x;
  int d = threadIdx.x;
  __shared__ float Bsh[16];
  __shared__ float Csh[16];
  float A[16], h[16];
#pragma unroll
  for (int n = 0; n < 16; ++n) {
    A[n] = -__expf(A_log[d * 16 + n]);
    h[n] = 0.f;
  }
  for (int t = 0; t < L; ++t) {
    long row = (long)b * L + t;
    if (d < 16) Bsh[d] = xdbl[row * 36 + 4 + d];
    else if (d < 32) Csh[d - 16] = xdbl[row * 36 + 20 + (d - 16)];
    __syncthreads();
    float dt = dtb[row * 128 + d];
    float xv = xc[row * 128 + d];
    float y = 0.f;
#pragma unroll
    for (int n = 0; n < 16; ++n) {
      h[n] = fmaf(h[n], __expf(dt * A[n]), dt * Bsh[n] * xv);
      y = fmaf(h[n], Csh[n], y);
    }
    ys[row * 128 + d] = y;
    __syncthreads();
  }
}

// y = (ys + D*xc) * silu(z)   (in place on ys)
__global__ void mamba_gate_kernel(float* __restrict__ ys,
                                  const float* __restrict__ xc,
                                  const float* __restrict__ xz,
                                  const float* __restrict__ Dp, long BL) {
  long gid = (long)blockIdx.x * blockDim.x + threadIdx.x;
  if (gid >= BL * 128) return;
  int d = gid & 127;
  long row = gid >> 7;
  float z = xz[row * 256 + 128 + d];
  float y = ys[gid] + Dp[d] * xc[gid];
  ys[gid] = y * (z * (1.0f / (1.0f + __expf(-z))));
}

// channel shuffle: out[j*g+gi] = in[gi*(C/g)+j]
__global__ void shuffle_kernel(const float* __restrict__ in,
                               float* __restrict__ out, int B, int C, int HW,
                               int g) {
  long gid = (long)blockIdx.x * blockDim.x + threadIdx.x;
  long total = (long)B * C * HW;
  if (gid >= total) return;
  int p = gid % HW;
  int oc = (gid / HW) % C;
  long b = gid / ((long)C * HW);
  int cg = C / g;
  int gi = oc % g, j = oc / g;
  out[gid] = in[(b * C + (long)gi * cg + j) * HW + p];
}

__global__ void zero_kernel(float* p, long n) {
  long gid = (long)blockIdx.x * blockDim.x + threadIdx.x;
  if (gid < n) p[gid] = 0.f;
}

// ---------------------------------------------------------------------------
// Host orchestration
// ---------------------------------------------------------------------------
struct MambaP {
  const float *in_w, *conv_w, *conv_b, *x_w, *dt_w, *dt_b, *A_log, *D, *out_w;
};

static void launch_conv(hipStream_t s, const float* in1, const float* in2,
                        int c_split, const float* w, const float* g,
                        const float* b, const float* resid, float* out, int B,
                        int Cin, int Cout, int H, int W, int ksz, int groups,
                        int act) {
  int HW = H * W;
  int strips = (HW + 31) / 32;  // 2 N-tiles per wave
  int mt = ((Cout / groups + 15) / 16) * groups;
  dim3 grid((strips + 3) / 4, mt, B);
  if (ksz == 3) {
    conv_wmma_kernel<3><<<grid, dim3(128), 0, s>>>(
        in1, in2, c_split, w, g, b, resid, out, B, Cin, Cout, H, W, groups,
        act);
  } else {
    conv_wmma_kernel<1><<<grid, dim3(128), 0, s>>>(
        in1, in2, c_split, w, g, b, resid, out, B, Cin, Cout, H, W, groups,
        act);
  }
}

static void launch_gemm(hipStream_t s, const float* A, const float* Bm,
                        float* out, int M, long N, int K, int ldb, int ldo,
                        const float* bias, int act) {
  long strips = (N + 31) / 32;  // 2 N-tiles per wave
  dim3 grid((unsigned)((strips + 3) / 4), (M + 15) / 16, 1);
  gemm_tn_wmma_kernel<<<grid, dim3(128), 0, s>>>(A, Bm, out, M, N, K, ldb, ldo,
                                                 bias, act);
}

static void run_mamba(hipStream_t s, const float* xflat, int Bseq, int L,
                      const MambaP& p, float* ybuf, float* XZ, float* XC,
                      float* XD, float* DT, float* YS) {
  long BL = (long)Bseq * L;
  launch_gemm(s, p.in_w, xflat, XZ, 256, BL, 64, 64, 256, nullptr, 0);
  {
    long total = BL * 128;
    dwconv1d_silu_kernel<<<(total + 255) / 256, 256, 0, s>>>(XZ, p.conv_w,
                                                             p.conv_b, XC, Bseq,
                                                             L);
  }
  launch_gemm(s, p.x_w, XC, XD, 36, BL, 128, 128, 36, nullptr, 0);
  launch_gemm(s, p.dt_w, XD, DT, 128, BL, 4, 36, 128, p.dt_b, 3);
  mamba_scan_kernel<<<Bseq, 128, 0, s>>>(DT, XC, XD, p.A_log, YS, L);
  {
    long total = BL * 128;
    mamba_gate_kernel<<<(total + 255) / 256, 256, 0, s>>>(YS, XC, XZ, p.D, BL);
  }
  launch_gemm(s, p.out_w, YS, ybuf, 64, BL, 128, 128, 64, nullptr, 0);
}

extern "C" void kernel_launch(void* const* d_in, const int* in_sizes, int n_in,
                              void* d_out, int out_size, void* d_ws,
                              size_t ws_size, hipStream_t stream) {
  auto F = [&](int i) { return (const float*)d_in[i]; };
  // Leaf order: setup_inputs() dict insertion order, params flattened
  // recursively in insertion order (89 leaves total).
  const float* f1c3 = F(0);  // 2*128*96*96
  const float* f1c2 = F(1);  // 2*128*48*48
  const float* f1c1 = F(2);  // 2*128*24*24
  const float* f2c3 = F(3);
  const float* f2c2 = F(4);
  const float* f2c1 = F(5);
  int i = 6;
  auto next = [&]() { return F(i++); };
  auto read_bc = [&](const float** w, const float** g, const float** b) {
    *w = next(); *g = next(); *b = next();  // _bc insertion order: w, g, b
  };
  auto read_mamba = [&](MambaP* m) {
    // _mamba_params insertion: in_w, conv_w, conv_b, x_w, dt_w, dt_b, A_log, D, out_w
    m->in_w = next(); m->conv_w = next(); m->conv_b = next(); m->x_w = next();
    m->dt_w = next(); m->dt_b = next(); m->A_log = next(); m->D = next();
    m->out_w = next();
  };
  struct CM { MambaP m1, m2; const float *cat_w, *cat_b; };
  auto read_cm = [&](CM* c) {
    read_mamba(&c->m1); read_mamba(&c->m2);
    c->cat_w = next(); c->cat_b = next();  // _cm insertion: m1, m2, cat_w, cat_b
  };
  const float *up1_w, *up1_g, *up1_b, *cat1_w, *cat1_g, *cat1_b;
  const float *up2w, *up2g, *up2b, *cat2_w, *cat2_g, *cat2_b;
  CM cm1, cm2, cm3;
  read_bc(&up1_w, &up1_g, &up1_b);    // up_conv1
  read_cm(&cm1);                      // mamba1
  read_bc(&cat1_w, &cat1_g, &cat1_b); // cat_conv1
  read_bc(&up2w, &up2g, &up2b);       // up_conv2
  read_cm(&cm2);                      // mamba2
  read_bc(&cat2_w, &cat2_g, &cat2_b); // cat_conv2
  read_cm(&cm3);                      // mamba3
  const float *c1w, *c1g, *c1b, *c2w, *c2g, *c2b, *c6w, *c6g, *c6b;
  read_bc(&c1w, &c1g, &c1b);          // decoder3.c1
  read_bc(&c2w, &c2g, &c2b);          // decoder3.c2
  read_bc(&c6w, &c6g, &c6b);          // decoder3.c6
  const float* c7w = next();
  const float* c7b = next();

  // ---- workspace arena ----
  float* W = (float*)d_ws;
  size_t o = 0;
  auto alloc = [&](size_t n) { float* p = W + o; o += n; return p; };
  float* bX0 = alloc((size_t)2 * 64 * 96 * 96);
  float* bX1 = alloc((size_t)2 * 64 * 96 * 96);
  float* bCV = alloc((size_t)2 * 52 * 96 * 96);
  float* bXZ = alloc((size_t)18432 * 256);
  float* bXC = alloc((size_t)18432 * 128);
  float* bXD = alloc((size_t)18432 * 36);
  float* bDT = alloc((size_t)18432 * 128);
  float* bYS = alloc((size_t)18432 * 128);
  float* bY1 = alloc((size_t)18432 * 64);
  float* bY2 = alloc((size_t)18432 * 64);
  float* bT1 = alloc((size_t)2 * 96 * 96 * 96);
  float* bT2 = alloc((size_t)2 * 96 * 96 * 96);
  if (o * sizeof(float) > ws_size) {  // insufficient scratch: deterministic out
    zero_kernel<<<(out_size + 255) / 256, 256, 0, stream>>>((float*)d_out,
                                                            out_size);
    return;
  }

  auto cross_mamba = [&](const float* x, float* out, int b, int c, int Hh,
                         int Ww, const CM& cm) {
    run_mamba(stream, x, b * Hh, Ww, cm.m1, bY1, bXZ, bXC, bXD, bDT, bYS);
    run_mamba(stream, x, b * Ww, Hh, cm.m2, bY2, bXZ, bXC, bXD, bDT, bYS);
    // 1x1 conv on concat(y1,y2) + cat_b + residual x
    launch_conv(stream, bY1, bY2, c, cm.cat_w, nullptr, cm.cat_b, x, out, b,
                2 * c, c, Hh, Ww, 1, 1, 0);
  };

  const int B = 2;
  // ---- level 1 (24x24 -> 48x48) ----
  {
    long tot = (long)B * 52 * 24 * 24;
    correlation_kernel<<<(tot + 255) / 256, 256, 0, stream>>>(f1c1, f2c1, bCV,
                                                              B, 128, 24, 24);
    long ut = (long)B * 52 * 48 * 48;
    up2_kernel<<<(ut + 255) / 256, 256, 0, stream>>>(bCV, bT1, B * 52, 24, 24);
  }
  launch_conv(stream, bT1, nullptr, 52, up1_w, up1_g, up1_b, nullptr, bX0, B,
              52, 64, 48, 48, 3, 1, /*softshrink*/ 2);
  cross_mamba(bX0, bX1, B, 64, 48, 48, cm1);

  // ---- level 2 (48x48 -> 96x96) ----
  {
    long tot = (long)B * 52 * 48 * 48;
    correlation_kernel<<<(tot + 255) / 256, 256, 0, stream>>>(f1c2, f2c2, bCV,
                                                              B, 128, 48, 48);
  }
  launch_conv(stream, bX1, bCV, 64, cat1_w, cat1_g, cat1_b, nullptr, bX0, B,
              116, 64, 48, 48, 3, 1, 2);
  {
    long ut = (long)B * 64 * 96 * 96;
    up2_kernel<<<(ut + 255) / 256, 256, 0, stream>>>(bX0, bT1, B * 64, 48, 48);
  }
  launch_conv(stream, bT1, nullptr, 64, up2w, up2g, up2b, nullptr, bX0, B, 64,
              64, 96, 96, 3, 1, 2);
  cross_mamba(bX0, bX1, B, 64, 96, 96, cm2);

  // ---- level 3 (96x96) ----
  {
    long tot = (long)B * 52 * 96 * 96;
    correlation_kernel<<<(tot + 255) / 256, 256, 0, stream>>>(f1c3, f2c3, bCV,
                                                              B, 128, 96, 96);
  }
  launch_conv(stream, bX1, bCV, 64, cat2_w, cat2_g, cat2_b, nullptr, bX0, B,
              116, 64, 96, 96, 3, 1, 2);
  cross_mamba(bX0, bX1, B, 64, 96, 96, cm3);

  // ---- light decoder ----
  const int HW = 96 * 96;
  launch_conv(stream, bX1, nullptr, 64, c1w, c1g, c1b, nullptr, bT1, B, 64, 96,
              96, 96, 3, 1, /*silu*/ 1);
  launch_conv(stream, bT1, nullptr, 96, c2w, c2g, c2b, nullptr, bT2, B, 96, 96,
              96, 96, 3, /*groups*/ 4, 1);
  {
    long tot = (long)B * 96 * HW;
    shuffle_kernel<<<(tot + 255) / 256, 256, 0, stream>>>(bT2, bT1, B, 96, HW,
                                                          4);
  }
  launch_conv(stream, bT1, nullptr, 96, c6w, c6g, c6b, nullptr, bT2, B, 96, 32,
              96, 96, 3, 1, 1);
  {
    long tot = (long)B * 32 * HW;
    shuffle_kernel<<<(tot + 255) / 256, 256, 0, stream>>>(bT2, bT1, B, 32, HW,
                                                          4);
  }
  launch_conv(stream, bT1, nullptr, 32, c7w, nullptr, c7b, nullptr,
              (float*)d_out, B, 32, 2, 96, 96, 3, 1, 0);
}